// SPO_snow_22127671509201
// MI455X (gfx1250) — compile-verified
//
#include <hip/hip_runtime.h>
#include <math.h>

typedef __attribute__((ext_vector_type(2))) float v2f;
typedef __attribute__((ext_vector_type(8))) float v8f;

#define N_PTS 4096
#define KNN   16
#define WAVES 4
#define BLOCK (WAVES * 32)

__global__ __launch_bounds__(BLOCK)
void knn_proj_loss_kernel(const float* __restrict__ pc_all, float* __restrict__ out)
{
    __shared__ float  sP[N_PTS * 3];          // per-batch cloud, stride-3 floats (48 KB)
    __shared__ float4 sTile[WAVES][32 * 4];   // per-wave WMMA staging, 16B-aligned (8 KB)
    __shared__ float  sRed;

    const int tid    = threadIdx.x;
    const int lane   = tid & 31;
    const int wv     = tid >> 5;
    const int b      = blockIdx.y;
    const int qBaseW = blockIdx.x * BLOCK + wv * 32;   // first query owned by this wave

    const float* pc = pc_all + (size_t)b * N_PTS * 3;

    if (tid == 0) sRed = 0.0f;

    // ---- Phase 1: stage the whole batch point cloud into LDS ----
    for (int i = tid; i < N_PTS; i += BLOCK) {
        sP[3 * i + 0] = pc[3 * i + 0];
        sP[3 * i + 1] = pc[3 * i + 1];
        sP[3 * i + 2] = pc[3 * i + 2];
    }
    __syncthreads();

    const int col = lane & 15;   // matrix column / row index within a 16-tile
    const int hv  = lane >> 4;   // which K-half this lane supplies (K={0,1} or {2,3})

    // A operands: rows (-2x, -2y, -2z, 1) for this wave's 32 queries (two 16-row tiles).
    // The -2 lives in A (built once) so B tiles (rebuilt 256x) stay multiply-free.
    v2f a0, a1;
    {
        const int q0 = qBaseW + col;
        const int q1 = qBaseW + 16 + col;
        const float x0 = sP[3 * q0 + 0], y0 = sP[3 * q0 + 1], z0 = sP[3 * q0 + 2];
        const float x1 = sP[3 * q1 + 0], y1 = sP[3 * q1 + 1], z1 = sP[3 * q1 + 2];
        a0.x = hv ? (-2.0f * z0) : (-2.0f * x0);
        a0.y = hv ? 1.0f         : (-2.0f * y0);
        a1.x = hv ? (-2.0f * z1) : (-2.0f * x1);
        a1.y = hv ? 1.0f         : (-2.0f * y1);
    }

    // Per-lane top-k state (sorted ascending, stable => lower index wins ties)
    float dk[KNN];
    int   ik[KNN];
#pragma unroll
    for (int j = 0; j < KNN; ++j) { dk[j] = 3.0e38f; ik[j] = 0; }

    float*        myTileF = (float*)&sTile[wv][0];
    const float4* myRow   = &sTile[wv][lane * 4];   // this lane's 16-value scan row

    // ---- KNN: sweep all candidate tiles with WMMA 16x16x4 ----
    for (int jt = 0; jt < N_PTS / 16; ++jt) {
        const int ci = jt * 16 + col;
        const float qx = sP[3 * ci + 0], qy = sP[3 * ci + 1], qz = sP[3 * ci + 2];
        const float sq = qx * qx + qy * qy + qz * qz;

        // B columns: (qx, qy, qz, |q|^2)  =>  D = |q|^2 - 2<p,q>
        v2f bb;
        bb.x = hv ? qz : qx;
        bb.y = hv ? sq : qy;

        v8f c0 = {};
        v8f d0 = __builtin_amdgcn_wmma_f32_16x16x4_f32(false, a0, false, bb,
                                                       (short)0, c0, false, false);
        v8f d1 = __builtin_amdgcn_wmma_f32_16x16x4_f32(false, a1, false, bb,
                                                       (short)0, c0, false, false);

        // Scatter D tiles into per-wave staging, row-major [32][16] floats.
        // (same-wave DS ops are in-order; region is private to this wave)
#pragma unroll
        for (int v = 0; v < 8; ++v) {
            const int r = v + 8 * hv;
            myTileF[r * 16 + col]        = d0[v];
            myTileF[(16 + r) * 16 + col] = d1[v];
        }

        // Vector-wide readback of this lane's 16 candidate values (4x ds_load_b128).
        const float4 r0 = myRow[0];
        const float4 r1 = myRow[1];
        const float4 r2 = myRow[2];
        const float4 r3 = myRow[3];
        const float vals[16] = { r0.x, r0.y, r0.z, r0.w,
                                 r1.x, r1.y, r1.z, r1.w,
                                 r2.x, r2.y, r2.z, r2.w,
                                 r3.x, r3.y, r3.z, r3.w };

        // Stable sorted top-16 maintenance.
        const int cBase = jt * 16;
#pragma unroll
        for (int c = 0; c < 16; ++c) {
            const float d = vals[c];
            if (d < dk[KNN - 1]) {
                const int cidx = cBase + c;
#pragma unroll
                for (int j = KNN - 1; j > 0; --j) {
                    const bool  sh = dk[j - 1] > d;   // old[j-1] shifts down
                    const float dj = dk[j];
                    const int   ij = ik[j];
                    dk[j] = sh ? dk[j - 1] : ((dj > d) ? d    : dj);
                    ik[j] = sh ? ik[j - 1] : ((dj > d) ? cidx : ij);
                }
                if (dk[0] > d) { dk[0] = d; ik[0] = cidx; }
            }
        }
    }

    // ---- Phase 2: per-query projection loss from its 16 NN ----
    const int   qi  = qBaseW + lane;
    const float px  = sP[3 * qi + 0], py = sP[3 * qi + 1], pz = sP[3 * qi + 2];
    const float sqp = px * px + py * py + pz * pz;

    float mx = 0.0f, my = 0.0f, mz = 0.0f;
#pragma unroll
    for (int c = 0; c < KNN; ++c) {
        const int n = ik[c];
        mx += sP[3 * n + 0]; my += sP[3 * n + 1]; mz += sP[3 * n + 2];
    }
    mx *= (1.0f / KNN); my *= (1.0f / KNN); mz *= (1.0f / KNN);
    const float mm = mx * mx + my * my + mz * mz;

    float acc = 0.0f;
#pragma unroll
    for (int c = 0; c < KNN; ++c) {
        const int   n   = ik[c];
        const float qx  = sP[3 * n + 0], qy = sP[3 * n + 1], qz = sP[3 * n + 2];
        const float sqq = qx * qx + qy * qy + qz * qz;
        const float dp  = dk[c] + sqp;                                     // |p-q|^2 (norm form)
        const float dm  = mm - 2.0f * (mx * qx + my * qy + mz * qz) + sqq; // |m-q|^2
        const float t   = (float)KNN * dp;
        acc += fabsf(dm - dp) / (t * t + 1.0f);
    }
    acc *= (1.0f / KNN);

    atomicAdd(&sRed, acc);                 // ds_add_f32
    __syncthreads();
    if (tid == 0) atomicAdd(out, sRed);    // global_atomic_add_f32
}

extern "C" void kernel_launch(void* const* d_in, const int* in_sizes, int n_in,
                              void* d_out, int out_size, void* d_ws, size_t ws_size,
                              hipStream_t stream)
{
    const float* pc = (const float*)d_in[0];
    float* out      = (float*)d_out;

    const int B = in_sizes[0] / (N_PTS * 3);   // = 8 for the reference shapes

    hipMemsetAsync(out, 0, sizeof(float) * (size_t)out_size, stream);

    dim3 grid(N_PTS / BLOCK, B);
    knn_proj_loss_kernel<<<grid, BLOCK, 0, stream>>>(pc, out);
}